// HashEncoderHyFluid_3530463117844
// MI455X (gfx1250) — compile-verified
//
#include <hip/hip_runtime.h>
#include <stdint.h>

#define NLEV 16

__global__ __launch_bounds__(256) void hash_encode_kernel(
    const float*    __restrict__ xyzts,      // [B,4]
    const float*    __restrict__ table,      // [total]
    const int*      __restrict__ shapes,     // [16,4]
    const uint32_t* __restrict__ sizes,      // [16]
    const int*      __restrict__ indicator,  // [16]
    const int*      __restrict__ offsets,    // [16]
    float*          __restrict__ out,        // [B, 32]
    int B)
{
    __shared__ float    s_resf[NLEV][4];
    __shared__ uint32_t s_stride[NLEV][4];
    __shared__ uint32_t s_size[NLEV];
    __shared__ uint32_t s_magic[NLEV];
    __shared__ uint32_t s_off2[NLEV];        // level offset in float2 units
    __shared__ int      s_ind[NLEV];
    __shared__ float    s_pts[64];           // 16 points x 4 coords

    const int tid       = threadIdx.x;
    const int pointBase = blockIdx.x * 16;

    // ---- stage per-level metadata into LDS (16 lanes, one per level) ----
    if (tid < NLEV) {
        const int r0 = shapes[tid * 4 + 0];
        const int r1 = shapes[tid * 4 + 1];
        const int r2 = shapes[tid * 4 + 2];
        const int r3 = shapes[tid * 4 + 3];
        s_resf[tid][0] = (float)r0;
        s_resf[tid][1] = (float)r1;
        s_resf[tid][2] = (float)r2;
        s_resf[tid][3] = (float)r3;
        const uint32_t a = (uint32_t)(r0 + 1);
        const uint32_t b = (uint32_t)(r1 + 1);
        const uint32_t c = (uint32_t)(r2 + 1);
        s_stride[tid][0] = 1u;
        s_stride[tid][1] = a;
        s_stride[tid][2] = a * b;
        s_stride[tid][3] = a * b * c;
        const uint32_t sz = sizes[tid];
        s_size[tid]  = sz;
        s_magic[tid] = 0xFFFFFFFFu / sz;     // Barrett magic: floor((2^32-1)/sz)
        s_off2[tid]  = ((uint32_t)offsets[tid]) >> 1;   // offsets are even
        s_ind[tid]   = indicator[tid];
    }
    // ---- stage the block's 16 points (64 floats) into LDS via the CDNA5
    //      async memory->LDS path: no VGPR round-trip, tracked by ASYNCcnt ----
    if (tid < 64) {
        const long long gi = (long long)pointBase * 4 + tid;
        if (gi < (long long)B * 4) {
            const float*   gp       = xyzts + gi;
            const uint32_t lds_addr = (uint32_t)(uintptr_t)&s_pts[tid]; // low 32b = LDS byte addr
            asm volatile("global_load_async_to_lds_b32 %0, %1, off"
                         :
                         : "v"(lds_addr), "v"(gp)
                         : "memory");
        }
    }
    asm volatile("s_wait_asynccnt 0x0" ::: "memory");
    __syncthreads();

    const int level  = tid & 15;
    const int plocal = tid >> 4;
    const int point  = pointBase + plocal;
    if (point >= B) return;

    // ---- per-dimension precomputation (corner-bit selectable tables) ----
    const uint32_t PR[4] = {1u, 2654435761u, 805459861u, 3674653429u};
    float    wsel[4][2];
    uint32_t cf[4][2];   // (grid+bit) * prime   (XOR-hash contributions)
    uint32_t cu[4][2];   // (grid+bit) * stride  (dense-index contributions)

#pragma unroll
    for (int d = 0; d < 4; ++d) {
        const float rf = s_resf[level][d];
        const float p  = s_pts[plocal * 4 + d] * rf;
        float gf = floorf(p);
        gf = fminf(fmaxf(gf, 0.0f), rf - 1.0f);
        float fr = p - gf;
        fr = fminf(fmaxf(fr, 0.0f), 1.0f);
        const uint32_t g  = (uint32_t)gf;
        const uint32_t st = s_stride[level][d];
        wsel[d][0] = 1.0f - fr;
        wsel[d][1] = fr;
        cf[d][0] = g * PR[d];
        cf[d][1] = (g + 1u) * PR[d];
        cu[d][0] = g * st;
        cu[d][1] = (g + 1u) * st;
    }

    // pair-factorized corner weights: w = wxy[c&3] * wzt[c>>2]
    float wxy[4], wzt[4];
#pragma unroll
    for (int j = 0; j < 2; ++j)
#pragma unroll
        for (int i = 0; i < 2; ++i) {
            wxy[(j << 1) | i] = wsel[0][i] * wsel[1][j];
            wzt[(j << 1) | i] = wsel[2][i] * wsel[3][j];
        }

    const uint32_t sz   = s_size[level];
    const uint32_t mg   = s_magic[level];
    const uint32_t off2 = s_off2[level];
    const int      ind  = s_ind[level];
    const float2* __restrict__ tp = (const float2*)table;

    // ---- phase 1: all 16 table indices (pure VALU) ----
    uint32_t fidx[16];
#pragma unroll
    for (int c = 0; c < 16; ++c) {
        const int b0 = c & 1, b1 = (c >> 1) & 1, b2 = (c >> 2) & 1, b3 = (c >> 3) & 1;
        const uint32_t h = ind ? (cu[0][b0] + cu[1][b1] + cu[2][b2] + cu[3][b3])
                               : (cf[0][b0] ^ cf[1][b1] ^ cf[2][b2] ^ cf[3][b3]);
        // Barrett reduction: r = h % sz  (magic => at most 2 fixups)
        const uint32_t q = (uint32_t)(((uint64_t)h * mg) >> 32);
        uint32_t r = h - q * sz;
        r = (r >= sz) ? r - sz : r;
        r = (r >= sz) ? r - sz : r;
        fidx[c] = off2 + r;                  // 32-bit float2 index into table
    }

    // ---- phase 2: issue all 16 gathers back-to-back (16-deep MLP) ----
    float2 v[16];
#pragma unroll
    for (int c = 0; c < 16; ++c)
        v[c] = tp[fidx[c]];                  // global_load_b64 saddr + 32-bit vindex

    // ---- phase 3: weighted accumulation (sequential order matches reference) ----
    float acc0 = 0.0f, acc1 = 0.0f;
#pragma unroll
    for (int c = 0; c < 16; ++c) {
        const float w = wxy[c & 3] * wzt[c >> 2];
        acc0 = fmaf(w, v[c].x, acc0);
        acc1 = fmaf(w, v[c].y, acc1);
    }

    // ---- packed non-temporal 8B store: streamed output must not evict the
    //      L2-resident hash table ----
    union { float f[2]; unsigned long long u; } pk;
    pk.f[0] = acc0;
    pk.f[1] = acc1;
    unsigned long long* dst =
        (unsigned long long*)(out + (size_t)point * (NLEV * 2) + level * 2);
    __builtin_nontemporal_store(pk.u, dst);
}

extern "C" void kernel_launch(void* const* d_in, const int* in_sizes, int n_in,
                              void* d_out, int out_size, void* d_ws, size_t ws_size,
                              hipStream_t stream) {
    const float*    xyzts     = (const float*)d_in[0];
    const float*    table     = (const float*)d_in[1];
    const int*      shapes    = (const int*)d_in[2];
    const uint32_t* sizes     = (const uint32_t*)d_in[3];
    const int*      indicator = (const int*)d_in[4];
    const int*      offsets   = (const int*)d_in[5];
    float*          out       = (float*)d_out;

    const int B = in_sizes[0] / 4;                 // xyzts is [B,4]
    const int threads = 256;                       // 8 wave32 waves, 16 points/block
    const int blocks  = (B * NLEV + threads - 1) / threads;

    hash_encode_kernel<<<blocks, threads, 0, stream>>>(
        xyzts, table, shapes, sizes, indicator, offsets, out, B);
}